// EMDLossWithProjection_17678085390414
// MI455X (gfx1250) — compile-verified
//
#include <hip/hip_runtime.h>
#include <hip/hip_bf16.h>
#include <math.h>

typedef __attribute__((ext_vector_type(2))) float v2f;
typedef __attribute__((ext_vector_type(8))) float v8f;
typedef __attribute__((ext_vector_type(4))) unsigned int u32x4;
typedef __attribute__((ext_vector_type(8))) int i32x8;
typedef __attribute__((ext_vector_type(4))) int i32x4;

#define EMD_EPS 1e-6f

// ---------------------------------------------------------------------------
// TDM helper: DMA a [tile_rows x 32] fp32 tile (row stride = row_stride_elems)
// from global into LDS at lds_off, padding +4 DWORDs after every 32 DWORDs so
// the LDS image has a 36-float row pitch (bank-conflict-free for WMMA frags).
//   D# group0: count=1 | lds_addr | global_addr(57b) | type=2
//   D# group1: data_size=4B, pad_enable, pad_interval=32dw, pad_amount=4dw,
//              tensor_dim0=stride, tensor_dim1=big, tile_dim0=32,
//              tile_dim1=rows, tensor_dim0_stride=stride
// Issued by one wave; tracked by TENSORcnt.
// ---------------------------------------------------------------------------
__device__ __forceinline__ void tdm_load_tile32(unsigned lds_off, const void* gsrc,
                                                int tile_rows, int row_stride_elems)
{
    unsigned long long ga = (unsigned long long)(size_t)gsrc;
    u32x4 g0;
    g0[0] = 1u;                                            // count = 1 (valid)
    g0[1] = lds_off;                                       // LDS byte address
    g0[2] = (unsigned)ga;                                  // global addr [31:0]
    g0[3] = (unsigned)((ga >> 32) & 0x1FFFFFFull) | (2u << 30); // [56:32] | type=2

    unsigned td0 = (unsigned)row_stride_elems;             // tensor_dim0
    unsigned td1 = 1u << 20;                               // tensor_dim1 (large)
    i32x8 g1;
    g1[0] = (int)((2u << 16) | (1u << 20) | (4u << 22) | (3u << 25));
    g1[1] = (int)((td0 & 0xFFFFu) << 16);                  // tensor_dim0[15:0]
    g1[2] = (int)((td0 >> 16) | ((td1 & 0xFFFFu) << 16));  // dim0[31:16] | dim1[15:0]
    g1[3] = (int)((td1 >> 16) | (32u << 16));              // dim1[31:16] | tile_dim0=32
    g1[4] = tile_rows;                                     // tile_dim1 (tile_dim2=0)
    g1[5] = row_stride_elems;                              // tensor_dim0_stride[31:0]
    g1[6] = 0;
    g1[7] = 0;

    i32x4 gz; gz[0] = 0; gz[1] = 0; gz[2] = 0; gz[3] = 0;
#if defined(__clang_major__) && (__clang_major__ >= 23)
    i32x8 gz8;
    gz8[0]=0; gz8[1]=0; gz8[2]=0; gz8[3]=0; gz8[4]=0; gz8[5]=0; gz8[6]=0; gz8[7]=0;
    __builtin_amdgcn_tensor_load_to_lds(g0, g1, gz, gz, gz8, 0);
#else
    __builtin_amdgcn_tensor_load_to_lds(g0, g1, gz, gz, 0);
#endif
}

// ---------------------------------------------------------------------------
// Kernel 1: aligned = S @ W^T   (C[m][n] = sum_k S[m,k] * W[n,k])
// 128x128 workgroup tile, 8 waves x (2x4 grid of 16x16 fp32 WMMA tiles).
// K slabs of 32 staged by the Tensor Data Mover, double-buffered so the DMA
// of slab k+1 overlaps the 64 WMMAs of slab k.
// ---------------------------------------------------------------------------
__global__ __launch_bounds__(256)
void emd_proj_gemm(const float* __restrict__ Smat, const float* __restrict__ Wmat,
                   float* __restrict__ Cmat, int M, int N, int K)
{
    __shared__ float As[2][128 * 36];   // [m][k] pitch 36 (TDM pad)
    __shared__ float Bs[2][128 * 36];   // [n][k] pitch 36 (rows of W)

    const int tid  = threadIdx.x;
    const int lane = tid & 31;
    const int wave = tid >> 5;
    const int m0 = blockIdx.y * 128;
    const int n0 = blockIdx.x * 128;

    const int wm = (wave & 3) * 32;    // wave M offset in tile
    const int wn = (wave >> 2) * 64;   // wave N offset in tile
    const int lr = lane & 15;          // M/N index within 16
    const int kh = (lane >> 4) * 2;    // K sub-offset: 0 or 2

    const bool issuer = (wave == 0);

    const v8f vz = {0.f,0.f,0.f,0.f,0.f,0.f,0.f,0.f};
    v8f acc[2][4];
    #pragma unroll
    for (int a = 0; a < 2; ++a)
      #pragma unroll
      for (int b = 0; b < 4; ++b) acc[a][b] = vz;

    // prime buffer 0
    if (issuer) {
        tdm_load_tile32((unsigned)(size_t)&As[0][0], &Smat[(size_t)m0 * K], 128, K);
        tdm_load_tile32((unsigned)(size_t)&Bs[0][0], &Wmat[(size_t)n0 * K], 128, K);
        __builtin_amdgcn_s_wait_tensorcnt(0);
    }
    __syncthreads();

    int buf = 0;
    for (int k0 = 0; k0 < K; k0 += 32) {
        if (issuer && (k0 + 32 < K)) {   // DMA next slab into the other buffer
            tdm_load_tile32((unsigned)(size_t)&As[buf ^ 1][0],
                            &Smat[(size_t)m0 * K + k0 + 32], 128, K);
            tdm_load_tile32((unsigned)(size_t)&Bs[buf ^ 1][0],
                            &Wmat[(size_t)n0 * K + k0 + 32], 128, K);
        }

        #pragma unroll
        for (int kk = 0; kk < 32; kk += 4) {
            v2f af[2], bf[4];
            #pragma unroll
            for (int mt = 0; mt < 2; ++mt) {
                float2 t = *reinterpret_cast<const float2*>(
                    &As[buf][(wm + mt * 16 + lr) * 36 + kk + kh]);
                af[mt][0] = t.x; af[mt][1] = t.y;
            }
            #pragma unroll
            for (int nt = 0; nt < 4; ++nt) {
                float2 t = *reinterpret_cast<const float2*>(
                    &Bs[buf][(wn + nt * 16 + lr) * 36 + kk + kh]);
                bf[nt][0] = t.x; bf[nt][1] = t.y;
            }
            #pragma unroll
            for (int mt = 0; mt < 2; ++mt)
              #pragma unroll
              for (int nt = 0; nt < 4; ++nt)
                acc[mt][nt] = __builtin_amdgcn_wmma_f32_16x16x4_f32(
                    false, af[mt], false, bf[nt], (short)0, acc[mt][nt], false, false);
        }

        if (issuer) __builtin_amdgcn_s_wait_tensorcnt(0);
        __syncthreads();
        buf ^= 1;
    }

    // C/D layout: VGPR r -> M = r + 8*(lane>>4), N = lane&15
    const int mrow = m0 + wm + 8 * (lane >> 4);
    const int ncol = n0 + wn + lr;
    #pragma unroll
    for (int mt = 0; mt < 2; ++mt)
      #pragma unroll
      for (int nt = 0; nt < 4; ++nt)
        #pragma unroll
        for (int r = 0; r < 8; ++r)
            Cmat[(size_t)(mrow + mt * 16 + r) * N + ncol + nt * 16] = acc[mt][nt][r];
}

// ---------------------------------------------------------------------------
// Kernel 2: mask[row] = (argmax_j aligned[row][j] != 0) ? 1 : 0
// (first-occurrence argmax, matching jnp.argmax)
// ---------------------------------------------------------------------------
__global__ __launch_bounds__(256)
void emd_argmax_mask(const float* __restrict__ A, float* __restrict__ mask, int Vd)
{
    __shared__ float sv[256];
    __shared__ int   si[256];
    const int row = blockIdx.x;
    const int tid = threadIdx.x;
    float best = -INFINITY; int bidx = 0x7fffffff;
    const float* r = &A[(size_t)row * Vd];
    for (int j = tid; j < Vd; j += 256) {
        float v = r[j];
        if (v > best) { best = v; bidx = j; }
    }
    sv[tid] = best; si[tid] = bidx;
    __syncthreads();
    for (int s = 128; s > 0; s >>= 1) {
        if (tid < s) {
            float ov = sv[tid + s]; int oi = si[tid + s];
            if (ov > sv[tid] || (ov == sv[tid] && oi < si[tid])) { sv[tid] = ov; si[tid] = oi; }
        }
        __syncthreads();
    }
    if (tid == 0) mask[row] = (si[0] != 0) ? 1.0f : 0.0f;
}

// ---------------------------------------------------------------------------
// Kernel 3: one workgroup per 64-row block.
//   s2/t2 row norms (vector loads), G = s@t^T via fp32 WMMA with TDM-staged
//   K slabs, D in LDS, then the flow/softmax/EMD math (64x64, tiny).
// ---------------------------------------------------------------------------
__global__ __launch_bounds__(256)
void emd_block_kernel(const float* __restrict__ A, const float* __restrict__ T,
                      const float* __restrict__ mask,
                      float* __restrict__ contribArr, float* __restrict__ validArr,
                      int Vd)
{
    __shared__ float Sa[64 * 36];
    __shared__ float Ta[64 * 36];
    __shared__ float Dm[64][65];
    __shared__ float s2[64], t2[64], marr[64], w0[64], wn[64], ct[64];
    __shared__ float red[256];
    __shared__ float scal[8];

    const int tid  = threadIdx.x;
    const int lane = tid & 31;
    const int wave = tid >> 5;
    const int rb   = blockIdx.x * 64;
    const float Vf = (float)Vd;

    if (tid < 64) marr[tid] = mask[rb + tid];

    // ---- phase 1: s2/t2 (4 threads per row, float4 loads) ----
    {
        int rrow = tid >> 2, part = tid & 3;
        const float* sp = &A[(size_t)(rb + rrow) * Vd];
        const float* tp = &T[(size_t)(rb + rrow) * Vd];
        float ss = 0.f, tt = 0.f;
        for (int k = part * 4; k < Vd; k += 16) {
            float4 a = *reinterpret_cast<const float4*>(&sp[k]);
            float4 b = *reinterpret_cast<const float4*>(&tp[k]);
            ss += a.x*a.x + a.y*a.y + a.z*a.z + a.w*a.w;
            tt += b.x*b.x + b.y*b.y + b.z*b.z + b.w*b.w;
        }
        red[tid] = ss;
        __syncthreads();
        if (part == 0) s2[rrow] = (red[tid] + red[tid+1] + red[tid+2] + red[tid+3]) / Vf;
        __syncthreads();
        red[tid] = tt;
        __syncthreads();
        if (part == 0) t2[rrow] = (red[tid] + red[tid+1] + red[tid+2] + red[tid+3]) / Vf;
        __syncthreads();
    }

    // ---- phase 2: G = s @ t^T with fp32 WMMA; TDM stages 64x32 slabs ----
    const int wm2 = (wave & 3) * 16;        // M tile
    const int wnb = (wave >> 2) * 32;       // two N tiles
    const int lr  = lane & 15;
    const int kh  = (lane >> 4) * 2;
    const bool issuer = (wave == 0);
    const v8f vz = {0.f,0.f,0.f,0.f,0.f,0.f,0.f,0.f};
    v8f acc[2]; acc[0] = vz; acc[1] = vz;

    for (int k0 = 0; k0 < Vd; k0 += 32) {
        if (issuer) {
            tdm_load_tile32((unsigned)(size_t)&Sa[0], &A[(size_t)rb * Vd + k0], 64, Vd);
            tdm_load_tile32((unsigned)(size_t)&Ta[0], &T[(size_t)rb * Vd + k0], 64, Vd);
            __builtin_amdgcn_s_wait_tensorcnt(0);
        }
        __syncthreads();
        #pragma unroll
        for (int kk = 0; kk < 32; kk += 4) {
            v2f af, bf[2];
            float2 ta = *reinterpret_cast<const float2*>(&Sa[(wm2 + lr) * 36 + kk + kh]);
            af[0] = ta.x; af[1] = ta.y;
            #pragma unroll
            for (int nt = 0; nt < 2; ++nt) {
                float2 tb = *reinterpret_cast<const float2*>(
                    &Ta[(wnb + nt * 16 + lr) * 36 + kk + kh]);
                bf[nt][0] = tb.x; bf[nt][1] = tb.y;
            }
            #pragma unroll
            for (int nt = 0; nt < 2; ++nt)
                acc[nt] = __builtin_amdgcn_wmma_f32_16x16x4_f32(
                    false, af, false, bf[nt], (short)0, acc[nt], false, false);
        }
        __syncthreads();
    }

    {   // D[i][j] = s2[i] + t2[j] - 2*G[i][j]/V
        const int mr = wm2 + 8 * (lane >> 4);
        #pragma unroll
        for (int nt = 0; nt < 2; ++nt)
          #pragma unroll
          for (int r = 0; r < 8; ++r) {
              int mm = mr + r;
              int nn = wnb + nt * 16 + lr;
              Dm[mm][nn] = s2[mm] + t2[nn] - 2.0f * acc[nt][r] / Vf;
          }
    }
    __syncthreads();

    // ---- phase 3: flow / softmax / EMD (64x64) ----
    if (tid == 0) {
        float ms = 0.f;
        for (int i = 0; i < 64; ++i) ms += marr[i];
        scal[0] = ms;                       // raw mask sum
        scal[1] = fmaxf(ms, EMD_EPS);       // clipped
    }
    __syncthreads();
    if (tid < 64) w0[tid] = (marr[tid] / 64.0f) / scal[1];
    __syncthreads();
    if (tid == 0) {
        float sw = 0.f;
        for (int i = 0; i < 64; ++i) sw += w0[i];
        float F0c = sw / 4096.0f;                         // min(sw,sw)/(B*B)
        float r1  = 1.0f / fmaxf(64.0f * F0c, EMD_EPS);   // row renorm
        float c1  = F0c * r1 * sw;                        // col sums of F1
        scal[2] = F0c * r1 / fmaxf(c1, EMD_EPS);          // F2 factor
    }
    __syncthreads();
    if (tid < 64) {                                       // C_T
        float fac = scal[2]; int i = tid; float s = 0.f;
        for (int j = 0; j < 64; ++j)
            s += Dm[i][j] * fmaxf(fac * w0[i] * w0[j], 0.f);
        ct[i] = s / fmaxf(w0[i], EMD_EPS);
    }
    __syncthreads();
    if (tid == 0) {
        float cs = 0.f;
        for (int i = 0; i < 64; ++i) cs += ct[i];
        scal[3] = cs;
    }
    __syncthreads();
    if (tid < 64) wn[tid] = scal[3] / fmaxf(ct[tid], EMD_EPS);   // w_bar
    __syncthreads();
    if (tid == 0) {
        float mx = -INFINITY;
        for (int i = 0; i < 64; ++i) mx = fmaxf(mx, wn[i]);
        scal[4] = mx;
    }
    __syncthreads();
    if (tid < 64) wn[tid] = __expf(wn[tid] - scal[4]);
    __syncthreads();
    if (tid == 0) {
        float es = 0.f;
        for (int i = 0; i < 64; ++i) es += wn[i];
        scal[5] = es;
    }
    __syncthreads();
    if (tid < 64) wn[tid] = wn[tid] / scal[5];                   // softmax
    __syncthreads();
    if (tid == 0) {
        float sw2 = 0.f;
        for (int i = 0; i < 64; ++i) sw2 += wn[i];
        float F0c = sw2 / 4096.0f;
        float r1  = 1.0f / fmaxf(64.0f * F0c, EMD_EPS);
        float c1  = F0c * r1 * sw2;
        scal[6] = F0c * r1 / fmaxf(c1, EMD_EPS);
    }
    __syncthreads();
    if (tid < 64) {                                       // emd num/denom
        float fac = scal[6]; int i = tid; float sn = 0.f, sd = 0.f;
        for (int j = 0; j < 64; ++j) {
            float f = fmaxf(fac * wn[i] * wn[j], 0.f);
            sn += f * Dm[i][j]; sd += f;
        }
        red[tid] = sn; red[64 + tid] = sd;
    }
    __syncthreads();
    if (tid == 0) {
        float sn = 0.f, sd = 0.f;
        for (int i = 0; i < 64; ++i) { sn += red[i]; sd += red[64 + i]; }
        float emd = sn / fmaxf(sd, EMD_EPS);
        float msr = scal[0];
        int valid = msr > 0.f;
        contribArr[blockIdx.x] = valid ? emd * (msr / 64.0f) : 0.0f;
        validArr[blockIdx.x]   = valid ? 1.0f : 0.0f;
    }
}

// ---------------------------------------------------------------------------
// Kernel 4: mean over valid blocks -> scalar
// ---------------------------------------------------------------------------
__global__ __launch_bounds__(256)
void emd_finalize(const float* __restrict__ contrib, const float* __restrict__ valid,
                  float* __restrict__ out, int nb)
{
    __shared__ float sc[256], sv[256];
    int tid = threadIdx.x;
    float c = 0.f, v = 0.f;
    for (int i = tid; i < nb; i += 256) { c += contrib[i]; v += valid[i]; }
    sc[tid] = c; sv[tid] = v;
    __syncthreads();
    for (int s = 128; s > 0; s >>= 1) {
        if (tid < s) { sc[tid] += sc[tid + s]; sv[tid] += sv[tid + s]; }
        __syncthreads();
    }
    if (tid == 0) out[0] = (sv[0] > 0.f) ? sc[0] / fmaxf(sv[0], 1.f) : 0.f;
}

// ---------------------------------------------------------------------------
extern "C" void kernel_launch(void* const* d_in, const int* in_sizes, int n_in,
                              void* d_out, int out_size, void* d_ws, size_t ws_size,
                              hipStream_t stream) {
    const float* S = (const float*)d_in[0];   // student_logits (Srows x V)
    const float* T = (const float*)d_in[1];   // teacher_logits (Srows x V)
    const float* W = (const float*)d_in[2];   // W (V x V)
    float* out = (float*)d_out;

    long nW = (long)in_sizes[2];
    int V = 1;
    while ((long)V * (long)V < nW) ++V;       // V = 8192
    int Srows = (int)((long)in_sizes[0] / V); // 8192
    int nb = Srows / 64;                      // 128 blocks

    float* aligned = (float*)d_ws;                       // Srows*V fp32
    float* mask    = aligned + (size_t)Srows * V;        // Srows
    float* contrib = mask + Srows;                       // nb
    float* valid   = contrib + nb;                       // nb

    dim3 g1(V / 128, Srows / 128);
    emd_proj_gemm<<<g1, dim3(256), 0, stream>>>(S, W, aligned, Srows, V, V);
    emd_argmax_mask<<<dim3(Srows), dim3(256), 0, stream>>>(aligned, mask, V);
    emd_block_kernel<<<dim3(nb), dim3(256), 0, stream>>>(aligned, T, mask, contrib, valid, V);
    emd_finalize<<<dim3(1), dim3(256), 0, stream>>>(contrib, valid, out, nb);
}